// Head_24120536335000
// MI455X (gfx1250) — compile-verified
//
#include <hip/hip_runtime.h>

typedef __attribute__((ext_vector_type(2))) float v2f;
typedef __attribute__((ext_vector_type(8))) float v8f;

constexpr int Bb = 8, Tt = 2048, Ee = 1024, Hh = 64;
constexpr int ROWS = Bb * Tt;            // 16384 flat rows
constexpr int TPB  = Tt / 16;            // 128 query tiles per batch

__device__ __forceinline__ v8f wmma4(v2f a, v2f b, v8f c) {
    // V_WMMA_F32_16X16X4_F32 : D = A(16x4) * B(4x16) + C(16x16)
    return __builtin_amdgcn_wmma_f32_16x16x4_f32(
        /*neg_a=*/false, a, /*neg_b=*/false, b,
        /*c_mod=*/(short)0, c, /*reuse_a=*/false, /*reuse_b=*/false);
}

// ---------------------------------------------------------------------------
// Kernel 1: K/Q/V projection.  grid = ROWS/16 blocks, 128 threads (4 waves).
// Wave w computes the 16-col H-tile w for one 16-row tile of x, for all three
// weight matrices simultaneously (A-operand of x is shared).
// ---------------------------------------------------------------------------
__global__ __launch_bounds__(128)
void proj_kernel(const float* __restrict__ x,
                 const float* __restrict__ Wk,
                 const float* __restrict__ Wq,
                 const float* __restrict__ Wv,
                 float* __restrict__ qo, float* __restrict__ ko,
                 float* __restrict__ vo) {
    const int lane   = threadIdx.x & 31;
    const int wv     = threadIdx.x >> 5;        // H-tile 0..3
    const int rowb   = blockIdx.x * 16;         // flat row base
    const int mrow   = lane & 15;
    const int khalf  = (lane >> 4) * 2;         // 0 or 2 (K sub-column)
    const int ncol   = wv * 16 + (lane & 15);

    const float* xa = x + (size_t)(rowb + mrow) * Ee + khalf;

    v8f cq = {}, ck = {}, cv = {};
    #pragma unroll 4
    for (int e0 = 0; e0 < Ee; e0 += 4) {
        v2f a = *(const v2f*)(xa + e0);                       // A[m][k..k+1]
        const float* wkp = Wk + (size_t)(e0 + khalf) * Hh + ncol;
        const float* wqp = Wq + (size_t)(e0 + khalf) * Hh + ncol;
        const float* wvp = Wv + (size_t)(e0 + khalf) * Hh + ncol;
        v2f bk; bk.x = wkp[0]; bk.y = wkp[Hh];                // B[k][n],B[k+1][n]
        v2f bq; bq.x = wqp[0]; bq.y = wqp[Hh];
        v2f bv; bv.x = wvp[0]; bv.y = wvp[Hh];
        ck = wmma4(a, bk, ck);
        cq = wmma4(a, bq, cq);
        cv = wmma4(a, bv, cv);
    }

    // C/D layout: row = r + 8*(lane>=16), col = lane&15
    const int orow = rowb + (lane >> 4) * 8;
    #pragma unroll
    for (int r = 0; r < 8; ++r) {
        size_t idx = (size_t)(orow + r) * Hh + ncol;
        qo[idx] = cq[r];
        ko[idx] = ck[r];
        vo[idx] = cv[r];
    }
}

// ---------------------------------------------------------------------------
// Kernel 2: flash attention, fp32 WMMA.  1 wave per 16-row query tile.
// ---------------------------------------------------------------------------
__global__ __launch_bounds__(32)
void attn_kernel(const float* __restrict__ q,
                 const float* __restrict__ k,
                 const float* __restrict__ v,
                 float* __restrict__ out) {
    __shared__ float pbuf[16][18];               // P tile, padded rows

    const int lane  = threadIdx.x & 31;
    const int tile  = blockIdx.x;                // 0..1023
    const int b     = tile / TPB;
    const int it    = tile % TPB;                // query tile within batch
    const int qrow0 = b * Tt + it * 16;          // flat row base of Q tile
    const int half  = lane >> 4;                 // 0 / 1
    const int nlo   = lane & 15;
    const int khalf = half * 2;

    // Preload the Q A-operand for all 16 k-steps (H = 64 = 16 * 4)
    v2f qa[16];
    const float* qp = q + (size_t)(qrow0 + nlo) * Hh + khalf;
    #pragma unroll
    for (int s = 0; s < 16; ++s) qa[s] = *(const v2f*)(qp + 4 * s);

    v8f o0 = {}, o1 = {}, o2 = {}, o3 = {};      // 16x64 output accumulator
    float mrow[8], lrow[8];
    #pragma unroll
    for (int r = 0; r < 8; ++r) { mrow[r] = -1e30f; lrow[r] = 0.f; }

    const float scale = 0.03125f;                // E^-0.5 = 1/32

    for (int j = 0; j <= it; ++j) {
        const int krow0 = b * Tt + j * 16;

        // ---- S = Q * K^T  (16x16), K-dim = 64 ----
        v8f s8 = {};
        const float* kp = k + (size_t)(krow0 + nlo) * Hh + khalf;
        #pragma unroll
        for (int t = 0; t < 16; ++t) {
            v2f kb = *(const v2f*)(kp + 4 * t);  // B[kk][n] = K[n-th keyrow][kk]
            s8 = wmma4(qa[t], kb, s8);
        }

        float sv[8];
        #pragma unroll
        for (int r = 0; r < 8; ++r) sv[r] = s8[r] * scale;

        if (j == it) {                           // causal mask on diagonal tile
            #pragma unroll
            for (int r = 0; r < 8; ++r) {
                int m = r + half * 8;
                if (nlo > m) sv[r] = -1e30f;
            }
        }

        // ---- online softmax (row reductions across 16-lane groups) ----
        float alpha[8], pv[8];
        #pragma unroll
        for (int r = 0; r < 8; ++r) {
            float t = sv[r];
            t = fmaxf(t, __shfl_xor(t, 1, 16));
            t = fmaxf(t, __shfl_xor(t, 2, 16));
            t = fmaxf(t, __shfl_xor(t, 4, 16));
            t = fmaxf(t, __shfl_xor(t, 8, 16));
            float mnew = fmaxf(mrow[r], t);
            alpha[r] = __expf(mrow[r] - mnew);
            mrow[r]  = mnew;
            float p  = __expf(sv[r] - mnew);
            pv[r] = p;
            float s = p;
            s += __shfl_xor(s, 1, 16);
            s += __shfl_xor(s, 2, 16);
            s += __shfl_xor(s, 4, 16);
            s += __shfl_xor(s, 8, 16);
            lrow[r] = lrow[r] * alpha[r] + s;
        }

        #pragma unroll
        for (int r = 0; r < 8; ++r) {
            o0[r] *= alpha[r]; o1[r] *= alpha[r];
            o2[r] *= alpha[r]; o3[r] *= alpha[r];
        }

        // ---- transpose P through LDS: C/D layout -> A-operand layout ----
        #pragma unroll
        for (int r = 0; r < 8; ++r)
            pbuf[r + half * 8][nlo] = pv[r];
        __builtin_amdgcn_wave_barrier();
        asm volatile("s_wait_dscnt 0" ::: "memory");

        // ---- O += P * V  (K-dim = 16) ----
        const float* vp = v + (size_t)krow0 * Hh;
        #pragma unroll
        for (int kk0 = 0; kk0 < 16; kk0 += 4) {
            v2f pa;
            pa.x = pbuf[nlo][kk0 + khalf];
            pa.y = pbuf[nlo][kk0 + khalf + 1];
            const float* vrow = vp + (size_t)(kk0 + khalf) * Hh;
            { v2f vb; vb.x = vrow[ 0 + nlo]; vb.y = vrow[Hh +  0 + nlo]; o0 = wmma4(pa, vb, o0); }
            { v2f vb; vb.x = vrow[16 + nlo]; vb.y = vrow[Hh + 16 + nlo]; o1 = wmma4(pa, vb, o1); }
            { v2f vb; vb.x = vrow[32 + nlo]; vb.y = vrow[Hh + 32 + nlo]; o2 = wmma4(pa, vb, o2); }
            { v2f vb; vb.x = vrow[48 + nlo]; vb.y = vrow[Hh + 48 + nlo]; o3 = wmma4(pa, vb, o3); }
        }
        __builtin_amdgcn_wave_barrier();
    }

    // ---- normalize and store ----
    #pragma unroll
    for (int r = 0; r < 8; ++r) {
        float inv = 1.0f / lrow[r];
        size_t idx = (size_t)(qrow0 + r + half * 8) * Hh;
        out[idx +  0 + nlo] = o0[r] * inv;
        out[idx + 16 + nlo] = o1[r] * inv;
        out[idx + 32 + nlo] = o2[r] * inv;
        out[idx + 48 + nlo] = o3[r] * inv;
    }
}

extern "C" void kernel_launch(void* const* d_in, const int* in_sizes, int n_in,
                              void* d_out, int out_size, void* d_ws, size_t ws_size,
                              hipStream_t stream) {
    const float* x  = (const float*)d_in[0];
    const float* Wk = (const float*)d_in[1];
    const float* Wq = (const float*)d_in[2];
    const float* Wv = (const float*)d_in[3];
    float* out = (float*)d_out;

    float* ws   = (float*)d_ws;
    float* qbuf = ws;
    float* kbuf = ws + (size_t)ROWS * Hh;
    float* vbuf = ws + 2 * (size_t)ROWS * Hh;   // 12 MB total fp32 scratch

    proj_kernel<<<ROWS / 16, 128, 0, stream>>>(x, Wk, Wq, Wv, qbuf, kbuf, vbuf);
    attn_kernel<<<ROWS / 16, 32, 0, stream>>>(qbuf, kbuf, vbuf, out);
}